// GCN_68728066670865
// MI455X (gfx1250) — compile-verified
//
#include <hip/hip_runtime.h>
#include <math.h>

// ---------------- problem constants (from reference) ----------------
#define NNODES 100000
#define NEDGES 1600000
#define NFEAT  256
#define NHID   128
#define NCLASS 40

typedef float v2f __attribute__((ext_vector_type(2)));
typedef float v8f __attribute__((ext_vector_type(8)));

// ---------------- degree / normalization ----------------
__global__ void init_deg_kernel(float* __restrict__ deg, int n) {
  int i = blockIdx.x * blockDim.x + threadIdx.x;
  if (i < n) deg[i] = 1.0f;  // self-loop contributes 1 to every node's degree
}

__global__ void deg_edges_kernel(const int* __restrict__ dst, float* __restrict__ deg, int e) {
  int i = blockIdx.x * blockDim.x + threadIdx.x;
  if (i < e) atomicAdd(&deg[dst[i]], 1.0f);
}

__global__ void dinv_kernel(const float* __restrict__ deg, float* __restrict__ dinv, int n) {
  int i = blockIdx.x * blockDim.x + threadIdx.x;
  if (i < n) dinv[i] = rsqrtf(deg[i]);  // deg >= 1 always (self loops)
}

// ---------------- fused WMMA f32 GEMM ----------------
// M[N,F]   = act(A)[N,K] @ W[K,F]           (act = bias+ReLU of previous layer if PRE)
// AGG[N,F] = M * dinv^2                     (self-loop term, fused epilogue)
// grid.x = N/16; block = 256 threads = 8 wave32 waves; wave w -> cols [16w,16w+16).
// A tile (16 x K) staged in LDS (float4, +4-float row pad: 16B aligned rows,
// bank stride 4 -> conflict-free b64 reads across the 16 lanes).
// V_WMMA_F32_16X16X4_F32 operand layouts (ISA 7.12.2):
//   A 16x4:  lane L holds M=L&15; VGPR0 = K=(L>=16?2:0), VGPR1 = K+1
//   B 4x16:  lane L holds N=L&15; VGPR0 = K=(L>=16?2:0), VGPR1 = K+1
//   C/D:     VGPR r, lane L -> M = r + (L>=16?8:0), N = L&15
// Out-of-range output columns (F=40 layer): clamp the B load address instead of
// predicating -- column n of D depends only on column n of B, so garbage lands
// only in lanes whose stores are skipped. No EXEC-mask branching in the loop.
template <int K, int F, bool PRE>
__global__ void __launch_bounds__(256) gemm_wmma_f32(const float* __restrict__ A,
                                                     const float* __restrict__ preBias,
                                                     const float* __restrict__ W,
                                                     const float* __restrict__ dinv,
                                                     float* __restrict__ M,
                                                     float* __restrict__ AGG) {
  constexpr int LDK = K + 4;
  __shared__ float As[16 * LDK];

  const int tid  = threadIdx.x;
  const int lane = tid & 31;
  const int wave = tid >> 5;
  const int row0 = blockIdx.x * 16;

  // cooperative float4 staging of the 16 x K tile (fused bias+ReLU when PRE)
  constexpr int K4 = K / 4;
  for (int idx = tid; idx < 16 * K4; idx += 256) {
    int r  = idx / K4;
    int c4 = idx % K4;
    float4 v = ((const float4*)(A + (size_t)(row0 + r) * K))[c4];
    if (PRE) {
      float4 bb = ((const float4*)preBias)[c4];
      v.x = fmaxf(v.x + bb.x, 0.0f);
      v.y = fmaxf(v.y + bb.y, 0.0f);
      v.z = fmaxf(v.z + bb.z, 0.0f);
      v.w = fmaxf(v.w + bb.w, 0.0f);
    }
    *(float4*)&As[r * LDK + c4 * 4] = v;
  }
  __syncthreads();

  constexpr int Fpad = (F + 15) & ~15;
  const int col0 = wave * 16;
  if (col0 >= Fpad) return;

  const int m    = lane & 15;
  const int koff = (lane >> 4) * 2;
  const int n    = col0 + m;
  const int nc   = (F % 16 == 0) ? n : (n < F ? n : F - 1);  // clamped load column

  v8f acc = {0.f, 0.f, 0.f, 0.f, 0.f, 0.f, 0.f, 0.f};
  const float* Wp = W + nc;

#pragma unroll 4
  for (int kk = 0; kk < K; kk += 4) {
    v2f a, b;
    a.x = As[m * LDK + kk + koff];
    a.y = As[m * LDK + kk + koff + 1];
    b.x = Wp[(size_t)(kk + koff) * F];
    b.y = Wp[(size_t)(kk + koff + 1) * F];
    acc = __builtin_amdgcn_wmma_f32_16x16x4_f32(false, a, false, b, (short)0, acc,
                                                false, false);
  }

  if (n < F) {
    const int rowadd = (lane >> 4) * 8;
#pragma unroll
    for (int r = 0; r < 8; ++r) {
      const int row = row0 + rowadd + r;
      const float di = dinv[row];
      const float v  = acc[r];
      M[(size_t)row * F + n]   = v;
      AGG[(size_t)row * F + n] = v * (di * di);
    }
  }
}

// ---------------- edge scatter-add: agg[dst] += m[src] * dinv[src]*dinv[dst] ----
// One wave per edge; float4 gathers (b128), f32 atomics; all L2-resident on
// MI455X (feature matrix is 51 MB << 192 MB L2).
__global__ void __launch_bounds__(256) edge_agg_kernel(const int* __restrict__ src,
                                                       const int* __restrict__ dst,
                                                       const float* __restrict__ dinv,
                                                       const float* __restrict__ m,
                                                       float* __restrict__ agg,
                                                       int E, int F4) {
  int e    = (blockIdx.x * blockDim.x + threadIdx.x) >> 5;
  int lane = threadIdx.x & 31;
  if (e >= E) return;
  int s = src[e];
  int d = dst[e];
  float w = dinv[s] * dinv[d];
  const float4* mr = (const float4*)m + (size_t)s * F4;
  float*        ar = agg + (size_t)d * (size_t)F4 * 4;
  for (int f = lane; f < F4; f += 32) {
    float4 v = mr[f];
    atomicAdd(&ar[4 * f + 0], v.x * w);
    atomicAdd(&ar[4 * f + 1], v.y * w);
    atomicAdd(&ar[4 * f + 2], v.z * w);
    atomicAdd(&ar[4 * f + 3], v.w * w);
  }
}

// ---------------- bias + ReLU + log_softmax, one wave per node ----------------
__global__ void __launch_bounds__(256) logsoftmax_kernel(const float* __restrict__ agg,
                                                         const float* __restrict__ b,
                                                         float* __restrict__ out,
                                                         int n, int F) {
  int node = (blockIdx.x * blockDim.x + threadIdx.x) >> 5;
  int lane = threadIdx.x & 31;
  if (node >= n) return;
  const float* row = agg + (size_t)node * F;

  int f1 = lane + 32;
  bool h0 = lane < F;
  bool h1 = f1 < F;
  float x0 = h0 ? fmaxf(row[lane] + b[lane], 0.0f) : 0.0f;
  float x1 = h1 ? fmaxf(row[f1] + b[f1], 0.0f) : 0.0f;

  float mv = h0 ? x0 : -INFINITY;
  if (h1) mv = fmaxf(mv, x1);
  for (int off = 16; off > 0; off >>= 1)
    mv = fmaxf(mv, __shfl_xor(mv, off, 32));

  float s = (h0 ? __expf(x0 - mv) : 0.0f) + (h1 ? __expf(x1 - mv) : 0.0f);
  for (int off = 16; off > 0; off >>= 1)
    s += __shfl_xor(s, off, 32);

  float lse = mv + __logf(s);
  float* orow = out + (size_t)node * F;
  if (h0) orow[lane] = x0 - lse;
  if (h1) orow[f1]   = x1 - lse;
}

// ---------------- driver ----------------
extern "C" void kernel_launch(void* const* d_in, const int* in_sizes, int n_in,
                              void* d_out, int out_size, void* d_ws, size_t ws_size,
                              hipStream_t stream) {
  const float* x  = (const float*)d_in[0];
  const int*   ei = (const int*)d_in[1];      // [2, E]: row0 = src, row1 = dst
  const float* W0 = (const float*)d_in[2];
  const float* b0 = (const float*)d_in[3];
  const float* W1 = (const float*)d_in[4];
  const float* b1 = (const float*)d_in[5];
  const float* W2 = (const float*)d_in[6];
  const float* b2 = (const float*)d_in[7];
  float* out = (float*)d_out;

  const int N = NNODES, E = NEDGES;
  const int* src = ei;
  const int* dst = ei + E;

  // workspace (floats): deg[N] | dinv[N] | m[N*128] | agg[N*128] | agg2[N*40]
  float* deg  = (float*)d_ws;
  float* dinv = deg + N;
  float* m    = dinv + N;
  float* agg  = m + (size_t)N * NHID;
  float* agg2 = agg + (size_t)N * NHID;

  const int T = 256;
  dim3 blk(T);

  // 1) degrees + normalization
  init_deg_kernel<<<(N + T - 1) / T, blk, 0, stream>>>(deg, N);
  deg_edges_kernel<<<(E + T - 1) / T, blk, 0, stream>>>(dst, deg, E);
  dinv_kernel<<<(N + T - 1) / T, blk, 0, stream>>>(deg, dinv, N);

  const int gemmBlocks = N / 16;            // 100000 / 16 = 6250 exactly
  const int edgeBlocks = (E * 32 + T - 1) / T;

  // 2) layer 0: m = x @ W0 ; agg = m*dinv^2 ; agg += scatter(m)
  gemm_wmma_f32<NFEAT, NHID, false><<<gemmBlocks, blk, 0, stream>>>(
      x, nullptr, W0, dinv, m, agg);
  edge_agg_kernel<<<edgeBlocks, blk, 0, stream>>>(src, dst, dinv, m, agg, E, NHID / 4);

  // 3) layer 1: A = relu(agg + b0) fused into A-tile staging; in-place agg ok (K==F)
  gemm_wmma_f32<NHID, NHID, true><<<gemmBlocks, blk, 0, stream>>>(
      agg, b0, W1, dinv, m, agg);
  edge_agg_kernel<<<edgeBlocks, blk, 0, stream>>>(src, dst, dinv, m, agg, E, NHID / 4);

  // 4) layer 2: F=40 -> separate output region (layout change breaks in-place)
  gemm_wmma_f32<NHID, NCLASS, true><<<gemmBlocks, blk, 0, stream>>>(
      agg, b1, W2, dinv, m, agg2);
  edge_agg_kernel<<<edgeBlocks, blk, 0, stream>>>(src, dst, dinv, m, agg2, E, NCLASS / 4);

  // 5) bias + ReLU + log_softmax
  logsoftmax_kernel<<<(N * 32 + T - 1) / T, blk, 0, stream>>>(agg2, b2, out, N, NCLASS);
}